// DeltaNet_22488448762180
// MI455X (gfx1250) — compile-verified
//
#include <hip/hip_runtime.h>
#include <math.h>

typedef float v2f __attribute__((ext_vector_type(2)));
typedef float v8f __attribute__((ext_vector_type(8)));

#define BB 2
#define TT 2048
#define CCH 1024
#define HHN 16
#define DDM 64
#define MMR (BB*TT)
#define EPSV 1e-6f
#define LN_EPSV 1e-5f

// ---------------------------------------------------------------------------
// GEMM: Y[M,N] = X[M,K] @ W[N,K]^T  (all fp32 row-major), exact fp32 via
// V_WMMA_F32_16X16X4_F32. One wave computes a 64x64 output tile:
// 4x4 grid of 16x16 accumulators, 16 WMMAs per 8 global loads per K-step.
// ---------------------------------------------------------------------------
__global__ __launch_bounds__(32)
void gemm_xwT_wmma(const float* __restrict__ X, const float* __restrict__ W,
                   float* __restrict__ Y, int Kdim, int Ndim)
{
    const int lane = threadIdx.x;           // 0..31
    const int m0 = blockIdx.y * 64;
    const int n0 = blockIdx.x * 64;
    const int rsel = lane & 15;             // row within 16-tile
    const int ksub = (lane >> 4) << 1;      // lanes 0-15 -> K0,K1 ; 16-31 -> K2,K3

    v8f acc[4][4];
    for (int i = 0; i < 4; ++i)
        for (int j = 0; j < 4; ++j)
            for (int r = 0; r < 8; ++r) acc[i][j][r] = 0.0f;

    const float* Xa[4];
    const float* Wb[4];
    for (int i = 0; i < 4; ++i) Xa[i] = X + (size_t)(m0 + i * 16 + rsel) * Kdim + ksub;
    for (int j = 0; j < 4; ++j) Wb[j] = W + (size_t)(n0 + j * 16 + rsel) * Kdim + ksub;

#pragma unroll 2
    for (int k0 = 0; k0 < Kdim; k0 += 4) {
        v2f a[4], b[4];
        for (int i = 0; i < 4; ++i) a[i] = *(const v2f*)(Xa[i] + k0);
        for (int j = 0; j < 4; ++j) b[j] = *(const v2f*)(Wb[j] + k0);
        for (int i = 0; i < 4; ++i)
            for (int j = 0; j < 4; ++j)
                acc[i][j] = __builtin_amdgcn_wmma_f32_16x16x4_f32(
                    false, a[i], false, b[j], (short)0, acc[i][j], false, false);
    }

    // C/D layout: VGPR r -> lanes 0-15: M=r, N=lane ; lanes 16-31: M=r+8, N=lane-16
    const int rofs = (lane >> 4) * 8;
    const int col  = lane & 15;
    for (int i = 0; i < 4; ++i)
        for (int j = 0; j < 4; ++j)
            for (int r = 0; r < 8; ++r) {
                int row = m0 + i * 16 + rofs + r;
                Y[(size_t)row * Ndim + n0 + j * 16 + col] = acc[i][j][r];
            }
}

// ---------------------------------------------------------------------------
// Gate: f[m,h] = clip(sigmoid(x[m,:]·Wg[h,:] + bg[h]), 0.01, 0.999)
// ---------------------------------------------------------------------------
__global__ __launch_bounds__(256)
void gate_kernel(const float* __restrict__ x, const float* __restrict__ Wg,
                 const float* __restrict__ bg, float* __restrict__ f)
{
    int gid = blockIdx.x * blockDim.x + threadIdx.x;
    if (gid >= MMR * HHN) return;
    int h = gid & (HHN - 1);
    int m = gid >> 4;
    const float* xr = x + (size_t)m * CCH;
    const float* wr = Wg + (size_t)h * CCH;
    float acc = 0.0f;
    for (int kk = 0; kk < CCH; ++kk) acc = fmaf(xr[kk], wr[kk], acc);
    acc += bg[h];
    float sg = 1.0f / (1.0f + expf(-acc));
    f[gid] = fminf(fmaxf(sg, 0.01f), 0.999f);
}

// ---------------------------------------------------------------------------
// RoPE + phi (elu+1) applied in place to q and k. One thread per (m,h,i<32).
// ---------------------------------------------------------------------------
__device__ __forceinline__ float phi_fn(float x) {
    return x > 0.0f ? x + 1.0f : expf(x);
}

__global__ __launch_bounds__(256)
void rope_phi_kernel(float* __restrict__ q, float* __restrict__ k)
{
    int gid = blockIdx.x * blockDim.x + threadIdx.x;
    if (gid >= MMR * HHN * 32) return;
    int i   = gid & 31;
    int rst = gid >> 5;
    int h   = rst & (HHN - 1);
    int mt  = rst >> 4;                 // token row in [0, M)
    int t   = mt & (TT - 1);            // position within sequence
    float inv_freq = powf(10000.0f, -(float)i * (1.0f / 32.0f));
    float ang = (float)t * inv_freq;
    float s, c;
    sincosf(ang, &s, &c);
    size_t base = ((size_t)mt * HHN + h) * DDM;

    float q1 = q[base + i], q2 = q[base + i + 32];
    q[base + i]      = phi_fn(q1 * c - q2 * s);
    q[base + i + 32] = phi_fn(q2 * c + q1 * s);

    float k1 = k[base + i], k2 = k[base + i + 32];
    k[base + i]      = phi_fn(k1 * c - k2 * s);
    k[base + i + 32] = phi_fn(k2 * c + k1 * s);
}

// ---------------------------------------------------------------------------
// Sequential gated scan. One block per (b,h); 64 threads (2 waves).
// Thread e owns state column S[:,e] (64 regs). Normalizer m kept redundantly
// per wave (2 elems/lane) so denom = q·m reduces with intra-wave shfl_xor --
// no barriers inside the per-step loop. q/k/v/f staged per 16-step chunk.
// ---------------------------------------------------------------------------
__global__ __launch_bounds__(64)
void scan_kernel(const float* __restrict__ q, const float* __restrict__ k,
                 const float* __restrict__ v, const float* __restrict__ f,
                 const int* __restrict__ mask, float* __restrict__ out)
{
    __shared__ float2 qk[16][64];   // (q,k) interleaved per step
    __shared__ float  vs[16][64];
    __shared__ float  fs[16];

    const int tid  = threadIdx.x;       // 0..63, owns column e = tid
    const int lane = tid & 31;
    const int bh = blockIdx.x;          // 0..31
    const int b  = bh >> 4;
    const int h  = bh & (HHN - 1);

    float S[64];
#pragma unroll
    for (int d = 0; d < 64; ++d) S[d] = 0.0f;
    float m_lo = 0.0f, m_hi = 0.0f;     // redundant per wave: m[lane], m[lane+32]

    for (int c = 0; c < TT / 16; ++c) {
        __syncthreads();                // previous chunk fully consumed
        for (int j = 0; j < 16; ++j) {
            int t = c * 16 + j;
            size_t row = ((size_t)(b * TT + t) * HHN + h) * DDM;
            float mk = (float)mask[b * TT + t];
            float qv = q[row + tid] * mk;
            float kv = k[row + tid] * mk;
            qk[j][tid] = make_float2(qv, kv);
            vs[j][tid] = v[row + tid] * mk;
            if (tid == j)
                fs[j] = f[(size_t)(b * TT + t) * HHN + h] * mk + (1.0f - mk);
        }
        __syncthreads();                // staging visible to both waves

        for (int j = 0; j < 16; ++j) {
            float ft = fs[j];
            float2 qlo = qk[j][lane];        // per-lane: q[lane],  k[lane]
            float2 qhi = qk[j][lane + 32];   // per-lane: q[l+32], k[l+32]
            m_lo = m_lo * ft + qlo.y;
            m_hi = m_hi * ft + qhi.y;
            float p = qlo.x * m_lo + qhi.x * m_hi;
#pragma unroll
            for (int off = 16; off > 0; off >>= 1)
                p += __shfl_xor(p, off, 32);
            float denom = fmaxf(p, EPSV);

            float ve = vs[j][tid];
            float accn = 0.0f;
#pragma unroll
            for (int d = 0; d < 64; ++d) {
                float2 qkd = qk[j][d];       // wave-uniform LDS broadcast
                S[d] = S[d] * ft + qkd.y * ve;
                accn = fmaf(qkd.x, S[d], accn);
            }
            int t = c * 16 + j;
            size_t row = ((size_t)(b * TT + t) * HHN + h) * DDM;
            out[row + tid] = accn / denom;
        }
    }
}

// ---------------------------------------------------------------------------
// Residual + bias + LayerNorm (two-pass, matches reference mean/var).
// One block per row, 256 threads x 4 columns.
// ---------------------------------------------------------------------------
__global__ __launch_bounds__(256)
void ln_kernel(const float* __restrict__ x, const float* __restrict__ proj,
               const float* __restrict__ bo, const float* __restrict__ lnw,
               const float* __restrict__ lnb, float* __restrict__ out)
{
    __shared__ float red[256];
    const int m = blockIdx.x;
    const int tid = threadIdx.x;

    float y[4];
    float s = 0.0f;
#pragma unroll
    for (int i = 0; i < 4; ++i) {
        int cdx = tid + i * 256;
        y[i] = x[(size_t)m * CCH + cdx] + proj[(size_t)m * CCH + cdx] + bo[cdx];
        s += y[i];
    }
    red[tid] = s;
    __syncthreads();
    for (int o = 128; o > 0; o >>= 1) {
        if (tid < o) red[tid] += red[tid + o];
        __syncthreads();
    }
    float mu = red[0] * (1.0f / CCH);
    __syncthreads();

    float s2 = 0.0f;
#pragma unroll
    for (int i = 0; i < 4; ++i) {
        float d = y[i] - mu;
        s2 += d * d;
    }
    red[tid] = s2;
    __syncthreads();
    for (int o = 128; o > 0; o >>= 1) {
        if (tid < o) red[tid] += red[tid + o];
        __syncthreads();
    }
    float var = red[0] * (1.0f / CCH);
    float inv = 1.0f / sqrtf(var + LN_EPSV);
#pragma unroll
    for (int i = 0; i < 4; ++i) {
        int cdx = tid + i * 256;
        out[(size_t)m * CCH + cdx] = (y[i] - mu) * inv * lnw[cdx] + lnb[cdx];
    }
}

// ---------------------------------------------------------------------------
extern "C" void kernel_launch(void* const* d_in, const int* in_sizes, int n_in,
                              void* d_out, int out_size, void* d_ws, size_t ws_size,
                              hipStream_t stream)
{
    (void)in_sizes; (void)n_in; (void)out_size; (void)ws_size;
    const float* x    = (const float*)d_in[0];
    const int*   mask = (const int*)d_in[1];
    const float* Wq   = (const float*)d_in[2];
    const float* Wk   = (const float*)d_in[3];
    const float* Wv   = (const float*)d_in[4];
    const float* Wg   = (const float*)d_in[5];
    const float* bg   = (const float*)d_in[6];
    const float* Wo   = (const float*)d_in[7];
    const float* bo   = (const float*)d_in[8];
    const float* lnw  = (const float*)d_in[9];
    const float* lnb  = (const float*)d_in[10];
    float* out = (float*)d_out;

    const size_t mc = (size_t)MMR * CCH;
    float* q    = (float*)d_ws;
    float* kbuf = q + mc;
    float* v    = kbuf + mc;
    float* attn = v + mc;
    float* proj = attn + mc;
    float* fbuf = proj + mc;

    dim3 gg(CCH / 64, MMR / 64);   // (16, 64) -> 1024 waves per GEMM
    gemm_xwT_wmma<<<gg, 32, 0, stream>>>(x, Wq, q,    CCH, CCH);
    gemm_xwT_wmma<<<gg, 32, 0, stream>>>(x, Wk, kbuf, CCH, CCH);
    gemm_xwT_wmma<<<gg, 32, 0, stream>>>(x, Wv, v,    CCH, CCH);
    gate_kernel<<<(MMR * HHN) / 256, 256, 0, stream>>>(x, Wg, bg, fbuf);
    rope_phi_kernel<<<(MMR * HHN * 32) / 256, 256, 0, stream>>>(q, kbuf);
    scan_kernel<<<BB * HHN, 64, 0, stream>>>(q, kbuf, v, fbuf, mask, attn);
    gemm_xwT_wmma<<<gg, 32, 0, stream>>>(attn, Wo, proj, CCH, CCH);
    ln_kernel<<<MMR, 256, 0, stream>>>(x, proj, bo, lnw, lnb, out);
}